// PoseMixtureVAE_up_858993459263
// MI455X (gfx1250) — compile-verified
//
#include <hip/hip_runtime.h>
#include <math.h>

typedef __attribute__((ext_vector_type(2))) float v2f;
typedef __attribute__((ext_vector_type(8))) float v8f;

#define WAVES_PER_BLOCK 8
#define BLOCK_THREADS 256
#define NUM_EXPERTS 16
#define MT 4   // M-tiles (16 rows) per wave; B fragment reused MT times

__device__ __forceinline__ float elu_act(float x) { return x > 0.f ? x : expm1f(x); }

// Packed weight layout: Wpk[(k>>2)*N*4 + col*4 + (k&3)]  (K padded to Kp%4==0,
// N padded to Np%16==0, zeros in padding). A lane reads its two B values
// (k+kh, k+kh+1 at column col) as ONE aligned b64 at  k*N + col*4 + kh.

// ---------------------------------------------------------------------------
// Branch-free, software-pipelined fp32 WMMA GEMM:
//   C = act(A[M,lda] @ Wpk(packed [Kp,N]) + bias[N])
// Kp%4==0, N%16==0, (M/16)%MT==0, A zero-padded to lda, Kp >= 8.
// ---------------------------------------------------------------------------
__global__ void wmma_gemm_kernel(const float* __restrict__ A,
                                 const float* __restrict__ W,   // packed
                                 const float* __restrict__ bias,
                                 float* __restrict__ C,
                                 int M, int N, int Kp, int lda, int ldc, int act)
{
    const int lane = threadIdx.x & 31;
    const int wave = threadIdx.x >> 5;
    const int ntN  = N >> 4;
    const int task = blockIdx.x * WAVES_PER_BLOCK + wave;
    const int total = (M >> 6) * ntN;              // MT==4 -> 64 rows per task
    if (task >= total) return;                     // wave-uniform: EXEC all-1s

    const int mG  = task / ntN;
    const int nT  = task % ntN;
    const int hi  = lane >> 4;
    const int l15 = lane & 15;
    const int kh  = hi << 1;                       // K sub-offset 0 or 2
    const int rowBase = (mG << 6) + l15;
    const int col     = (nT << 4) + l15;

    const float* __restrict__ Ap = A + (size_t)rowBase * lda + kh;
    const float* __restrict__ Wp = W + (size_t)col * 4 + kh;

    v8f acc[MT] = {};
    v2f a[MT], b;
    b = *reinterpret_cast<const v2f*>(Wp);
#pragma unroll
    for (int t = 0; t < MT; ++t)
        a[t] = *reinterpret_cast<const v2f*>(Ap + (size_t)t * 16 * lda);

    for (int k = 0; k < Kp - 4; k += 4) {
        Ap += 4;
        Wp += (size_t)4 * N;
        v2f bn = *reinterpret_cast<const v2f*>(Wp);        // prefetch k+4
        v2f an[MT];
#pragma unroll
        for (int t = 0; t < MT; ++t)
            an[t] = *reinterpret_cast<const v2f*>(Ap + (size_t)t * 16 * lda);
#pragma unroll
        for (int t = 0; t < MT; ++t)                        // compute k
            acc[t] = __builtin_amdgcn_wmma_f32_16x16x4_f32(false, a[t], false, b,
                                                           (short)0, acc[t], false, false);
        b = bn;
#pragma unroll
        for (int t = 0; t < MT; ++t) a[t] = an[t];
    }
#pragma unroll
    for (int t = 0; t < MT; ++t)                            // last K-step
        acc[t] = __builtin_amdgcn_wmma_f32_16x16x4_f32(false, a[t], false, b,
                                                       (short)0, acc[t], false, false);

    const float bv = bias ? bias[col] : 0.f;
    const int voff = hi ? 8 : 0;
#pragma unroll
    for (int t = 0; t < MT; ++t) {
        const int mBase = (mG << 6) + t * 16 + voff;
#pragma unroll
        for (int v = 0; v < 8; ++v) {
            float val = acc[t][v] + bv;
            if (act) val = elu_act(val);
            C[(size_t)(mBase + v) * ldc + col] = val;
        }
    }
}

// ---------------------------------------------------------------------------
// Fused soft-MoE GEMM: C[b,o] = act( sum_e coef[b,e] * ((A@W[e])[b,o] + Wb[e,o]) )
// W: [E][Kp,Np] packed; Wb: [E,Np] zero-padded; coef: [M,E]. Store masked on
// col < Nout (ldc = Nout). Expert loop fused in-register.
// ---------------------------------------------------------------------------
__global__ void wmma_moe_kernel(const float* __restrict__ A,
                                const float* __restrict__ W,   // packed per expert
                                const float* __restrict__ Wb,
                                const float* __restrict__ coef,
                                float* __restrict__ C,
                                int M, int Np, int Nout, int Kp, int lda, int ldc, int act)
{
    const int lane = threadIdx.x & 31;
    const int wave = threadIdx.x >> 5;
    const int ntN  = Np >> 4;
    const int task = blockIdx.x * WAVES_PER_BLOCK + wave;
    const int total = (M >> 6) * ntN;
    if (task >= total) return;                     // wave-uniform

    const int mG  = task / ntN;
    const int nT  = task % ntN;
    const int hi  = lane >> 4;
    const int l15 = lane & 15;
    const int kh  = hi << 1;
    const int rowBase = (mG << 6) + l15;
    const int col     = (nT << 4) + l15;
    const int voff    = hi ? 8 : 0;

    const float* __restrict__ Abase = A + (size_t)rowBase * lda + kh;

    v8f facc[MT] = {};
    for (int e = 0; e < NUM_EXPERTS; ++e) {
        const float* __restrict__ Ap = Abase;
        const float* __restrict__ Wp = W + (size_t)e * Kp * Np + (size_t)col * 4 + kh;

        v8f acc[MT] = {};
        v2f a[MT], b;
        b = *reinterpret_cast<const v2f*>(Wp);
#pragma unroll
        for (int t = 0; t < MT; ++t)
            a[t] = *reinterpret_cast<const v2f*>(Ap + (size_t)t * 16 * lda);

        for (int k = 0; k < Kp - 4; k += 4) {
            Ap += 4;
            Wp += (size_t)4 * Np;
            v2f bn = *reinterpret_cast<const v2f*>(Wp);
            v2f an[MT];
#pragma unroll
            for (int t = 0; t < MT; ++t)
                an[t] = *reinterpret_cast<const v2f*>(Ap + (size_t)t * 16 * lda);
#pragma unroll
            for (int t = 0; t < MT; ++t)
                acc[t] = __builtin_amdgcn_wmma_f32_16x16x4_f32(false, a[t], false, b,
                                                               (short)0, acc[t], false, false);
            b = bn;
#pragma unroll
            for (int t = 0; t < MT; ++t) a[t] = an[t];
        }
#pragma unroll
        for (int t = 0; t < MT; ++t)
            acc[t] = __builtin_amdgcn_wmma_f32_16x16x4_f32(false, a[t], false, b,
                                                           (short)0, acc[t], false, false);

        const float be = Wb[(size_t)e * Np + col];
#pragma unroll
        for (int t = 0; t < MT; ++t) {
            const int mBase = (mG << 6) + t * 16 + voff;
#pragma unroll
            for (int v = 0; v < 8; ++v) {
                const float cf = coef[(size_t)(mBase + v) * NUM_EXPERTS + e];
                facc[t][v] += cf * (acc[t][v] + be);
            }
        }
    }

    if (col < Nout) {                               // epilogue-only divergence
#pragma unroll
        for (int t = 0; t < MT; ++t) {
            const int mBase = (mG << 6) + t * 16 + voff;
#pragma unroll
            for (int v = 0; v < 8; ++v) {
                float val = facc[t][v];
                if (act) val = elu_act(val);
                C[(size_t)(mBase + v) * ldc + col] = val;
            }
        }
    }
}

// ---------------------------------------------------------------------------
// out[b, 0:n1] = A1[b,:]; out[b, n1:n1+n2] = A2[b,:]; zero-pad to ldo
// ---------------------------------------------------------------------------
__global__ void concat2_kernel(const float* __restrict__ A1, int n1,
                               const float* __restrict__ A2, int n2,
                               float* __restrict__ out, int ldo, int M)
{
    const int idx = blockIdx.x * blockDim.x + threadIdx.x;
    const int total = M * ldo;
    if (idx >= total) return;
    const int b = idx / ldo, c = idx % ldo;
    float v = 0.f;
    if (c < n1)            v = A1[(size_t)b * n1 + c];
    else if (c < n1 + n2)  v = A2[(size_t)b * n2 + (c - n1)];
    out[idx] = v;
}

// Pad [K,N] -> [Kp,Np] with zeros AND pack into WMMA-B layout
// out[b][(k>>2)*Np*4 + n*4 + (k&3)]; batched over blockIdx.y.
__global__ void pack_w_kernel(const float* __restrict__ in, float* __restrict__ out,
                              int K, int N, int Kp, int Np)
{
    const int b = blockIdx.y;
    const int idx = blockIdx.x * blockDim.x + threadIdx.x;
    if (idx >= Kp * Np) return;
    const int k = idx / Np, n = idx % Np;
    float v = 0.f;
    if (k < K && n < N) v = in[(size_t)b * K * N + (size_t)k * N + n];
    out[(size_t)b * Kp * Np + (size_t)(k >> 2) * Np * 4 + (size_t)n * 4 + (k & 3)] = v;
}

// Zero-padded plain 2D copy (for biases): out[kp,np] = in[k,n] or 0
__global__ void pad2d_kernel(const float* __restrict__ in, float* __restrict__ out,
                             int K, int N, int Kp, int Np)
{
    const int idx = blockIdx.x * blockDim.x + threadIdx.x;
    if (idx >= Kp * Np) return;
    const int kp = idx / Np, np = idx % Np;
    float v = 0.f;
    if (kp < K && np < N) v = in[(size_t)kp * N + np];
    out[idx] = v;
}

// z = mu + eps * exp(0.5*logvar)
__global__ void reparam_kernel(const float* __restrict__ mu, const float* __restrict__ lv,
                               const float* __restrict__ eps, float* __restrict__ z, int n)
{
    const int i = blockIdx.x * blockDim.x + threadIdx.x;
    if (i >= n) return;
    z[i] = mu[i] + eps[i] * expf(0.5f * lv[i]);
}

// row-wise softmax over E=16 logits
__global__ void softmax16_kernel(const float* __restrict__ logits,
                                 float* __restrict__ coef, int M)
{
    const int b = blockIdx.x * blockDim.x + threadIdx.x;
    if (b >= M) return;
    float v[NUM_EXPERTS];
    float mx = -INFINITY;
#pragma unroll
    for (int i = 0; i < NUM_EXPERTS; ++i) { v[i] = logits[(size_t)b * NUM_EXPERTS + i]; mx = fmaxf(mx, v[i]); }
    float s = 0.f;
#pragma unroll
    for (int i = 0; i < NUM_EXPERTS; ++i) { v[i] = expf(v[i] - mx); s += v[i]; }
    const float inv = 1.f / s;
#pragma unroll
    for (int i = 0; i < NUM_EXPERTS; ++i) coef[(size_t)b * NUM_EXPERTS + i] = v[i] * inv;
}

// ---------------------------------------------------------------------------
extern "C" void kernel_launch(void* const* d_in, const int* in_sizes, int n_in,
                              void* d_out, int out_size, void* d_ws, size_t ws_size,
                              hipStream_t stream)
{
    (void)in_sizes; (void)n_in; (void)out_size; (void)ws_size;
    constexpr int B = 2048, FC = 171, FR = 171, L = 32, H = 256, E = 16, GH = 64;
    constexpr int IN = L + FC;    // 203
    constexpr int INT_ = L + H;   // 288
    constexpr int OUT = FR;       // 171
    constexpr int OUTP = 176;     // 171 padded to multiple of 16

    const float* x     = (const float*)d_in[0];
    const float* c     = (const float*)d_in[1];
    const float* eps   = (const float*)d_in[2];
    const float* fc1_w = (const float*)d_in[3];  const float* fc1_b = (const float*)d_in[4];
    const float* fc2_w = (const float*)d_in[5];  const float* fc2_b = (const float*)d_in[6];
    const float* mu_w  = (const float*)d_in[7];  const float* mu_b  = (const float*)d_in[8];
    const float* lv_w  = (const float*)d_in[9];  const float* lv_b  = (const float*)d_in[10];
    const float* g0_w  = (const float*)d_in[11]; const float* g0_b  = (const float*)d_in[12];
    const float* g1_w  = (const float*)d_in[13]; const float* g1_b  = (const float*)d_in[14];
    const float* g2_w  = (const float*)d_in[15]; const float* g2_b  = (const float*)d_in[16];
    const float* w0    = (const float*)d_in[17]; const float* b0    = (const float*)d_in[18];
    const float* w1    = (const float*)d_in[19]; const float* b1    = (const float*)d_in[20];
    const float* w2    = (const float*)d_in[21]; const float* b2    = (const float*)d_in[22];

    float* out   = (float*)d_out;                 // [B, OUT]
    float* muO   = out + (size_t)B * OUT;         // [B, L]
    float* lvO   = muO + (size_t)B * L;           // [B, L]

    // ---- workspace layout (fp32) ----
    float* ws = (float*)d_ws;
    float* cat1 = ws;  ws += (size_t)B * 344;     // [x,c]   K 342 -> 344
    float* h1   = ws;  ws += (size_t)B * H;
    float* cat2 = ws;  ws += (size_t)B * 428;     // [x,h1]  K 427 -> 428
    float* h2   = ws;  ws += (size_t)B * H;
    float* cat3 = ws;  ws += (size_t)B * 428;     // [x,h2]
    float* z    = ws;  ws += (size_t)B * L;
    float* gcat = ws;  ws += (size_t)B * 204;     // [z,c]   K 203 -> 204
    float* g1b  = ws;  ws += (size_t)B * GH;
    float* g2b  = ws;  ws += (size_t)B * GH;
    float* glog = ws;  ws += (size_t)B * E;
    float* coef = ws;  ws += (size_t)B * E;
    float* l0   = ws;  ws += (size_t)B * H;
    float* inp1 = ws;  ws += (size_t)B * INT_;    // [z,l0]  K=288
    float* l1   = ws;  ws += (size_t)B * H;
    float* inp2 = ws;  ws += (size_t)B * INT_;    // [z,l1]
    // packed (padded) weights
    float* fc1_wp = ws; ws += (size_t)344 * H;
    float* fc2_wp = ws; ws += (size_t)428 * H;
    float* mu_wp  = ws; ws += (size_t)428 * L;
    float* lv_wp  = ws; ws += (size_t)428 * L;
    float* g0_wp  = ws; ws += (size_t)204 * GH;
    float* g1_wp  = ws; ws += (size_t)GH * GH;
    float* g2_wp  = ws; ws += (size_t)GH * E;
    float* w0p    = ws; ws += (size_t)E * 204 * H;
    float* w1p    = ws; ws += (size_t)E * INT_ * H;
    float* w2p    = ws; ws += (size_t)E * INT_ * OUTP;
    float* b2p    = ws; ws += (size_t)E * OUTP;

    auto gemm_blocks = [](int M, int N) {
        int tasks = (M >> 6) * (N >> 4);
        return (tasks + WAVES_PER_BLOCK - 1) / WAVES_PER_BLOCK;
    };
    auto ew_blocks = [](long n) { return (int)((n + BLOCK_THREADS - 1) / BLOCK_THREADS); };

    // ---- one-time weight pad + pack into WMMA-B layout ----
    pack_w_kernel<<<dim3(ew_blocks((long)344 * H), 1), BLOCK_THREADS, 0, stream>>>(fc1_w, fc1_wp, 342, H, 344, H);
    pack_w_kernel<<<dim3(ew_blocks((long)428 * H), 1), BLOCK_THREADS, 0, stream>>>(fc2_w, fc2_wp, 427, H, 428, H);
    pack_w_kernel<<<dim3(ew_blocks((long)428 * L), 1), BLOCK_THREADS, 0, stream>>>(mu_w, mu_wp, 427, L, 428, L);
    pack_w_kernel<<<dim3(ew_blocks((long)428 * L), 1), BLOCK_THREADS, 0, stream>>>(lv_w, lv_wp, 427, L, 428, L);
    pack_w_kernel<<<dim3(ew_blocks((long)204 * GH), 1), BLOCK_THREADS, 0, stream>>>(g0_w, g0_wp, 203, GH, 204, GH);
    pack_w_kernel<<<dim3(ew_blocks((long)GH * GH), 1), BLOCK_THREADS, 0, stream>>>(g1_w, g1_wp, GH, GH, GH, GH);
    pack_w_kernel<<<dim3(ew_blocks((long)GH * E), 1), BLOCK_THREADS, 0, stream>>>(g2_w, g2_wp, GH, E, GH, E);
    pack_w_kernel<<<dim3(ew_blocks((long)204 * H), E), BLOCK_THREADS, 0, stream>>>(w0, w0p, IN, H, 204, H);
    pack_w_kernel<<<dim3(ew_blocks((long)INT_ * H), E), BLOCK_THREADS, 0, stream>>>(w1, w1p, INT_, H, INT_, H);
    pack_w_kernel<<<dim3(ew_blocks((long)INT_ * OUTP), E), BLOCK_THREADS, 0, stream>>>(w2, w2p, INT_, OUT, INT_, OUTP);
    pad2d_kernel<<<ew_blocks((long)E * OUTP), BLOCK_THREADS, 0, stream>>>(b2, b2p, E, OUT, E, OUTP);

    // ---- Encoder ----
    concat2_kernel<<<ew_blocks((long)B * 344), BLOCK_THREADS, 0, stream>>>(x, FR, c, FC, cat1, 344, B);
    wmma_gemm_kernel<<<gemm_blocks(B, H), BLOCK_THREADS, 0, stream>>>(
        cat1, fc1_wp, fc1_b, h1, B, H, 344, 344, H, 1);
    concat2_kernel<<<ew_blocks((long)B * 428), BLOCK_THREADS, 0, stream>>>(x, FR, h1, H, cat2, 428, B);
    wmma_gemm_kernel<<<gemm_blocks(B, H), BLOCK_THREADS, 0, stream>>>(
        cat2, fc2_wp, fc2_b, h2, B, H, 428, 428, H, 1);
    concat2_kernel<<<ew_blocks((long)B * 428), BLOCK_THREADS, 0, stream>>>(x, FR, h2, H, cat3, 428, B);
    wmma_gemm_kernel<<<gemm_blocks(B, L), BLOCK_THREADS, 0, stream>>>(
        cat3, mu_wp, mu_b, muO, B, L, 428, 428, L, 0);
    wmma_gemm_kernel<<<gemm_blocks(B, L), BLOCK_THREADS, 0, stream>>>(
        cat3, lv_wp, lv_b, lvO, B, L, 428, 428, L, 0);
    reparam_kernel<<<ew_blocks((long)B * L), BLOCK_THREADS, 0, stream>>>(muO, lvO, eps, z, B * L);

    // ---- Gate MLP + softmax coefficients ----
    concat2_kernel<<<ew_blocks((long)B * 204), BLOCK_THREADS, 0, stream>>>(z, L, c, FC, gcat, 204, B);
    wmma_gemm_kernel<<<gemm_blocks(B, GH), BLOCK_THREADS, 0, stream>>>(
        gcat, g0_wp, g0_b, g1b, B, GH, 204, 204, GH, 1);
    wmma_gemm_kernel<<<gemm_blocks(B, GH), BLOCK_THREADS, 0, stream>>>(
        g1b, g1_wp, g1_b, g2b, B, GH, GH, GH, GH, 1);
    wmma_gemm_kernel<<<gemm_blocks(B, E), BLOCK_THREADS, 0, stream>>>(
        g2b, g2_wp, g2_b, glog, B, E, GH, GH, E, 0);
    softmax16_kernel<<<ew_blocks(B), BLOCK_THREADS, 0, stream>>>(glog, coef, B);

    // ---- Soft-MoE decoder (expert loop fused in-register) ----
    wmma_moe_kernel<<<gemm_blocks(B, H), BLOCK_THREADS, 0, stream>>>(
        gcat, w0p, b0, coef, l0, B, H, H, 204, 204, H, 1);
    concat2_kernel<<<ew_blocks((long)B * INT_), BLOCK_THREADS, 0, stream>>>(z, L, l0, H, inp1, INT_, B);
    wmma_moe_kernel<<<gemm_blocks(B, H), BLOCK_THREADS, 0, stream>>>(
        inp1, w1p, b1, coef, l1, B, H, H, INT_, INT_, H, 1);
    concat2_kernel<<<ew_blocks((long)B * INT_), BLOCK_THREADS, 0, stream>>>(z, L, l1, H, inp2, INT_, B);
    wmma_moe_kernel<<<gemm_blocks(B, OUTP), BLOCK_THREADS, 0, stream>>>(
        inp2, w2p, b2p, coef, out, B, OUTP, OUT, INT_, INT_, OUT, 0);
}